// GraphNet_7902739825285
// MI455X (gfx1250) — compile-verified
//
#include <hip/hip_runtime.h>
#include <math.h>

// ---- problem constants (from reference setup_inputs) ----
#define Bn 16
#define Cn 512
#define Nn 4096              // H*W = 64*64
#define Kn 64
#define FLATSZ (Bn*Kn*Cn)    // 524288 floats: first output
#define EPSR 1e-9f

typedef __attribute__((ext_vector_type(2))) float v2f;
typedef __attribute__((ext_vector_type(8))) float v8f;
typedef int v4i_ __attribute__((vector_size(16)));   // pointee type the async builtin expects

#define AS1 __attribute__((address_space(1)))
#define AS3 __attribute__((address_space(3)))

#if __has_builtin(__builtin_amdgcn_global_load_async_to_lds_b128)
#define ASYNC_OK 1
#else
#define ASYNC_OK 0
#endif

// one 16-byte async DMA transfer global -> LDS (per lane)
__device__ __forceinline__ void async_copy16(const void* g, void* l) {
#if ASYNC_OK
  __builtin_amdgcn_global_load_async_to_lds_b128(
      (AS1 v4i_*)(size_t)g,   // global ptrs are bit-identical generic vs AS1
      (AS3 v4i_*)l, 0, 0);
#endif
}

template <int N>
__device__ __forceinline__ void wait_async() {
#if ASYNC_OK
#if __has_builtin(__builtin_amdgcn_s_wait_asynccnt)
  __builtin_amdgcn_s_wait_asynccnt(N);
#else
  asm volatile("s_wait_asynccnt %0" ::"i"(N) : "memory");
#endif
#endif
}

// ------------------------------------------------------------------
// Kernel 1: parameter prep.
//   sigma = sigmoid(raw); constk[k] = sum_c a^2/sigma^2
//   Wpack[cp][m][k][v]  (cp=c>>1, v=c&1, m=0: 1/s^2, m=1: -2a/s^2)
//   -> every distance-GEMM B-fragment becomes one coalesced b64 load.
// ------------------------------------------------------------------
__global__ __launch_bounds__(Cn)
void prep_kernel(const float* __restrict__ anchor,
                 const float* __restrict__ sigma_raw,
                 float* __restrict__ wpack, float* __restrict__ sigma,
                 float* __restrict__ constk) {
  const int k = blockIdx.x;
  const int c = threadIdx.x;
  const int idx = k * Cn + c;
  const float a = anchor[idx];
  const float s = 1.0f / (1.0f + __expf(-sigma_raw[idx]));
  const float is2 = 1.0f / (s * s);
  sigma[idx] = s;
  const int cp = c >> 1, v = c & 1;
  wpack[(cp * 2 + 0) * (Kn * 2) + k * 2 + v] = is2;
  wpack[(cp * 2 + 1) * (Kn * 2) + k * 2 + v] = -2.0f * a * is2;

  __shared__ float red[Cn];
  red[c] = a * a * is2;
  __syncthreads();
  for (int off = Cn / 2; off > 0; off >>= 1) {
    if (c < off) red[c] += red[c + off];
    __syncthreads();
  }
  if (c == 0) constk[k] = red[0];
}

// ------------------------------------------------------------------
// Kernel 2: distance GEMM + softmax over K, writes soft_assign [B,K,N].
// Block = 128 threads (4 waves), covers 128 pixels of one batch.
// Wave: 32 pixels x 64 nodes (2 M-tiles x 4 N-tiles = 8 f32 accumulators).
// x staged to LDS with double-buffered GLOBAL_LOAD_ASYNC_TO_LDS_B128.
// ------------------------------------------------------------------
#define XS 136        // LDS row stride for x chunk [32][XS] (bank-safe)
#define SAS 132       // LDS row stride for soft_assign transpose [64][SAS]

__global__ __launch_bounds__(128)
void dist_softmax_kernel(const float* __restrict__ x,
                         const float* __restrict__ wpack,
                         const float* __restrict__ constk,
                         float* __restrict__ sa /* [B,K,N] */) {
  const int b      = blockIdx.x >> 5;            // 32 blocks per batch
  const int nblock = (blockIdx.x & 31) * 128;
  const int tid  = threadIdx.x;
  const int wave = tid >> 5;
  const int lane = tid & 31;
  const int l    = lane & 15;
  const int h    = lane >> 4;

  __shared__ float lds[2 * 32 * XS];             // 2 x-chunk buffers (34.8KB)

  const float* xb = x + (size_t)b * Cn * Nn + nblock;

  // stage x[b, cb..cb+31, nblock..nblock+127]: 1024 float4, 8 per thread
  auto stage = [&](float* dst, int cb) {
#pragma unroll
    for (int i = 0; i < 8; i++) {
      const int q  = tid + i * 128;
      const int r  = q >> 5;         // row 0..31
      const int c4 = q & 31;         // float4 col 0..31
      const float* g = xb + (size_t)(cb + r) * Nn + c4 * 4;
      float* ds = dst + r * XS + c4 * 4;
#if ASYNC_OK
      async_copy16(g, ds);
#else
      *(float4*)ds = *(const float4*)g;
#endif
    }
  };

  v8f acc[2][4];
#pragma unroll
  for (int mt = 0; mt < 2; mt++)
#pragma unroll
    for (int t = 0; t < 4; t++) acc[mt][t] = (v8f){0,0,0,0,0,0,0,0};

  const int colA0 = wave * 32 + l;   // pixel cols for the two A tiles
  const int colA1 = colA0 + 16;

  stage(lds, 0);
  for (int cb = 0; cb < Cn; cb += 32) {
    const int par = (cb >> 5) & 1;
    float* cur = lds + par * (32 * XS);
    float* nxt = lds + (1 - par) * (32 * XS);
    if (cb + 32 < Cn) {
      stage(nxt, cb + 32);
      wait_async<8>();               // cur's 8 transfers done, nxt in flight
    } else {
      wait_async<0>();
    }
    __syncthreads();

#pragma unroll
    for (int s = 0; s < 8; s++) {
      const int rr = 4 * s + 2 * h;
      const float a00 = cur[(rr + 0) * XS + colA0];
      const float a01 = cur[(rr + 1) * XS + colA0];
      const float a10 = cur[(rr + 0) * XS + colA1];
      const float a11 = cur[(rr + 1) * XS + colA1];
      v2f aX0 = {a00, a01}, aS0 = {a00 * a00, a01 * a01};
      v2f aX1 = {a10, a11}, aS1 = {a10 * a10, a11 * a11};

      const int cp0 = ((cb + 4 * s) >> 1) + h;
      const float* w0 = wpack + (size_t)cp0 * 2 * (Kn * 2);  // mat0 row
      const float* w1 = w0 + (Kn * 2);                       // mat1 row
#pragma unroll
      for (int t = 0; t < 4; t++) {
        v2f b1 = *(const v2f*)(w0 + (16 * t + l) * 2);
        v2f b2 = *(const v2f*)(w1 + (16 * t + l) * 2);
        acc[0][t] = __builtin_amdgcn_wmma_f32_16x16x4_f32(false, aS0, false, b1, (short)0, acc[0][t], false, false);
        acc[0][t] = __builtin_amdgcn_wmma_f32_16x16x4_f32(false, aX0, false, b2, (short)0, acc[0][t], false, false);
        acc[1][t] = __builtin_amdgcn_wmma_f32_16x16x4_f32(false, aS1, false, b1, (short)0, acc[1][t], false, false);
        acc[1][t] = __builtin_amdgcn_wmma_f32_16x16x4_f32(false, aX1, false, b2, (short)0, acc[1][t], false, false);
      }
    }
    __syncthreads();   // all waves done reading cur before it is re-staged
  }

  // logits = -0.5*(d2+const); softmax over K in place (4 local + shfl 16)
  const float ck[4] = {constk[l], constk[16 + l], constk[32 + l], constk[48 + l]};
#pragma unroll
  for (int mt = 0; mt < 2; mt++) {
#pragma unroll
    for (int vr = 0; vr < 8; vr++) {
      float l0 = -0.5f * (acc[mt][0][vr] + ck[0]);
      float l1 = -0.5f * (acc[mt][1][vr] + ck[1]);
      float l2 = -0.5f * (acc[mt][2][vr] + ck[2]);
      float l3 = -0.5f * (acc[mt][3][vr] + ck[3]);
      float mx = fmaxf(fmaxf(l0, l1), fmaxf(l2, l3));
#pragma unroll
      for (int m = 1; m < 16; m <<= 1) mx = fmaxf(mx, __shfl_xor(mx, m, 32));
      float e0 = __expf(l0 - mx), e1 = __expf(l1 - mx);
      float e2 = __expf(l2 - mx), e3 = __expf(l3 - mx);
      float sm = (e0 + e1) + (e2 + e3);
#pragma unroll
      for (int m = 1; m < 16; m <<= 1) sm += __shfl_xor(sm, m, 32);
      const float inv = 1.0f / sm;
      acc[mt][0][vr] = e0 * inv; acc[mt][1][vr] = e1 * inv;
      acc[mt][2][vr] = e2 * inv; acc[mt][3][vr] = e3 * inv;
    }
  }

  // transpose through LDS -> coalesced [K][n] stores (last barrier above
  // guarantees no wave is still reading the x buffers)
#pragma unroll
  for (int mt = 0; mt < 2; mt++)
#pragma unroll
    for (int t = 0; t < 4; t++)
#pragma unroll
      for (int vr = 0; vr < 8; vr++)
        lds[(16 * t + l) * SAS + wave * 32 + mt * 16 + vr + 8 * h] = acc[mt][t][vr];
  __syncthreads();

  float* sab = sa + (size_t)b * Kn * Nn + nblock;
#pragma unroll
  for (int i = 0; i < 16; i++) {
    const int q  = tid + i * 128;   // 2048 float4 = 64x128 floats
    const int r  = q >> 5;
    const int c4 = q & 31;
    float4 vv = *(const float4*)(lds + r * SAS + c4 * 4);
    *(float4*)(sab + (size_t)r * Nn + c4 * 4) = vv;
  }
}

// ------------------------------------------------------------------
// Kernel 3: w_sum[b,k] = sum_n soft_assign[b,k,n]
// ------------------------------------------------------------------
__global__ __launch_bounds__(256)
void wsum_kernel(const float* __restrict__ sa, float* __restrict__ wsum) {
  const int bk = blockIdx.x;
  const int tid = threadIdx.x;
  const float* p = sa + (size_t)bk * Nn;
  float s = 0.0f;
  for (int i = tid; i < Nn; i += 256) s += p[i];
  __shared__ float red[256];
  red[tid] = s;
  __syncthreads();
  for (int off = 128; off > 0; off >>= 1) {
    if (tid < off) red[tid] += red[tid + off];
    __syncthreads();
  }
  if (tid == 0) wsum[bk] = red[0];
}

// ------------------------------------------------------------------
// Kernel 4: aggregation GEMM  wx[b,k,c] = sum_n S[b,k,n] * x[b,c,n]
// Block = 256 threads (8 waves). Wave (seg, kh): 32k x 64c over a
// quarter of the N-reduce; partials combined through a 64KB LDS tile.
// Grid = B*8 -> 1024 waves.
// ------------------------------------------------------------------
__global__ __launch_bounds__(256)
void agg_kernel(const float* __restrict__ sa, const float* __restrict__ x,
                float* __restrict__ wx) {
  const int b    = blockIdx.x >> 3;
  const int cblk = (blockIdx.x & 7) * 64;
  const int tid  = threadIdx.x;
  const int wave = tid >> 5;
  const int lane = tid & 31;
  const int l = lane & 15, h = lane >> 4;
  const int kh  = wave & 1;         // which 32-k half
  const int seg = wave >> 1;        // n-segment 0..3
  const int n0  = seg * (Nn / 4);

  v8f acc[2][4];
#pragma unroll
  for (int kt = 0; kt < 2; kt++)
#pragma unroll
    for (int t = 0; t < 4; t++) acc[kt][t] = (v8f){0,0,0,0,0,0,0,0};

  const float* sb0 = sa + ((size_t)b * Kn + kh * 32 + l) * Nn + n0;
  const float* sb1 = sb0 + (size_t)16 * Nn;
  const float* xb  = x + (size_t)b * Cn * Nn + n0;
  const float* xr0 = xb + (size_t)(cblk +  0 + l) * Nn;
  const float* xr1 = xb + (size_t)(cblk + 16 + l) * Nn;
  const float* xr2 = xb + (size_t)(cblk + 32 + l) * Nn;
  const float* xr3 = xb + (size_t)(cblk + 48 + l) * Nn;

#pragma unroll 2
  for (int nc = 0; nc < Nn / 4; nc += 4) {
    const int off = nc + 2 * h;
    v2f a0 = *(const v2f*)(sb0 + off);
    v2f a1 = *(const v2f*)(sb1 + off);
    v2f b0 = *(const v2f*)(xr0 + off);
    v2f b1 = *(const v2f*)(xr1 + off);
    v2f b2 = *(const v2f*)(xr2 + off);
    v2f b3 = *(const v2f*)(xr3 + off);
    acc[0][0] = __builtin_amdgcn_wmma_f32_16x16x4_f32(false, a0, false, b0, (short)0, acc[0][0], false, false);
    acc[0][1] = __builtin_amdgcn_wmma_f32_16x16x4_f32(false, a0, false, b1, (short)0, acc[0][1], false, false);
    acc[0][2] = __builtin_amdgcn_wmma_f32_16x16x4_f32(false, a0, false, b2, (short)0, acc[0][2], false, false);
    acc[0][3] = __builtin_amdgcn_wmma_f32_16x16x4_f32(false, a0, false, b3, (short)0, acc[0][3], false, false);
    acc[1][0] = __builtin_amdgcn_wmma_f32_16x16x4_f32(false, a1, false, b0, (short)0, acc[1][0], false, false);
    acc[1][1] = __builtin_amdgcn_wmma_f32_16x16x4_f32(false, a1, false, b1, (short)0, acc[1][1], false, false);
    acc[1][2] = __builtin_amdgcn_wmma_f32_16x16x4_f32(false, a1, false, b2, (short)0, acc[1][2], false, false);
    acc[1][3] = __builtin_amdgcn_wmma_f32_16x16x4_f32(false, a1, false, b3, (short)0, acc[1][3], false, false);
  }

  // combine the 4 n-segment partials through LDS
  __shared__ float red[8 * 2048];   // 64KB: [wave][32k x 64c]
  float* my = red + wave * 2048;
#pragma unroll
  for (int kt = 0; kt < 2; kt++)
#pragma unroll
    for (int t = 0; t < 4; t++)
#pragma unroll
      for (int vr = 0; vr < 8; vr++)
        my[(kt * 16 + vr + 8 * h) * 64 + t * 16 + l] = acc[kt][t][vr];
  __syncthreads();

  float* wxb = wx + (size_t)b * Kn * Cn + cblk;
#pragma unroll
  for (int i = 0; i < 16; i++) {
    const int o  = tid + i * 256;   // 4096 outputs (64k x 64c)
    const int kl = o >> 6;
    const int cl = o & 63;
    const int khh = kl >> 5, kin = kl & 31;
    const int base = khh * 2048 + kin * 64 + cl;
    float s = red[base] + red[base + 2 * 2048] +
              red[base + 4 * 2048] + red[base + 6 * 2048];
    wxb[(size_t)kl * Cn + cl] = s;
  }
}

// ------------------------------------------------------------------
// Kernel 5: nodes = (wx - wsum*a)/sigma/(wsum+eps), per-node L2 norm
// in place; record post-normalization sum of squares.
// ------------------------------------------------------------------
__global__ __launch_bounds__(256)
void node_norm_kernel(float* __restrict__ wx, const float* __restrict__ wsum,
                      const float* __restrict__ anchor,
                      const float* __restrict__ sigma,
                      float* __restrict__ postsq) {
  const int bk = blockIdx.x;
  const int k  = bk & (Kn - 1);
  const int tid = threadIdx.x;
  const float ws = wsum[bk];
  const float inv_wse = 1.0f / (ws + EPSR);
  float* p = wx + (size_t)bk * Cn;

  float v[2];
  float ss = 0.0f;
#pragma unroll
  for (int i = 0; i < 2; i++) {
    const int c = tid + i * 256;
    const float val = (p[c] - ws * anchor[k * Cn + c]) / sigma[k * Cn + c] * inv_wse;
    v[i] = val;
    ss += val * val;
  }
  __shared__ float red[256];
  red[tid] = ss;
  __syncthreads();
  for (int off = 128; off > 0; off >>= 1) {
    if (tid < off) red[tid] += red[tid + off];
    __syncthreads();
  }
  const float tot = red[0];
  const float mn  = fmaxf(sqrtf(tot), 1e-12f);
  const float inv = 1.0f / mn;
#pragma unroll
  for (int i = 0; i < 2; i++) p[tid + i * 256] = v[i] * inv;
  if (tid == 0) postsq[bk] = tot * inv * inv;
}

// ------------------------------------------------------------------
// Kernel 6: global L2 over each batch row, write first output.
// ------------------------------------------------------------------
__global__ __launch_bounds__(256)
void global_norm_kernel(const float* __restrict__ wx,
                        const float* __restrict__ postsq,
                        float* __restrict__ out) {
  const int b = blockIdx.x;
  const int tid = threadIdx.x;
  __shared__ float red[64];
  if (tid < 64) red[tid] = postsq[b * Kn + tid];
  __syncthreads();
  for (int off = 32; off > 0; off >>= 1) {
    if (tid < off) red[tid] += red[tid + off];
    __syncthreads();
  }
  const float inv = 1.0f / fmaxf(sqrtf(red[0]), 1e-12f);
  const float* src = wx + (size_t)b * Kn * Cn;
  float* dst = out + (size_t)b * Kn * Cn;
  for (int i = tid; i < Kn * Cn; i += 256) dst[i] = src[i] * inv;
}

// ------------------------------------------------------------------
extern "C" void kernel_launch(void* const* d_in, const int* in_sizes, int n_in,
                              void* d_out, int out_size, void* d_ws, size_t ws_size,
                              hipStream_t stream) {
  const float* x         = (const float*)d_in[0];  // [B,C,H,W]
  const float* anchor    = (const float*)d_in[1];  // [K,C]
  const float* sigma_raw = (const float*)d_in[2];  // [K,C]
  float* out = (float*)d_out;
  float* ws  = (float*)d_ws;

  // workspace layout (floats, ~2.5 MB total)
  float* wpack  = ws;             // 65536  (packed WMMA-ready weights)
  float* sigma  = ws + 65536;     // 32768
  float* constk = ws + 98304;     // 64
  float* wsum   = ws + 98368;     // 1024
  float* wx     = ws + 99392;     // 524288
  float* postsq = ws + 623680;    // 1024

  float* sa = out + FLATSZ;       // soft_assign written straight into output

  prep_kernel<<<Kn, Cn, 0, stream>>>(anchor, sigma_raw, wpack, sigma, constk);
  dist_softmax_kernel<<<Bn * 32, 128, 0, stream>>>(x, wpack, constk, sa);
  wsum_kernel<<<Bn * Kn, 256, 0, stream>>>(sa, wsum);
  agg_kernel<<<Bn * 8, 256, 0, stream>>>(sa, x, wx);
  node_norm_kernel<<<Bn * Kn, 256, 0, stream>>>(wx, wsum, anchor, sigma, postsq);
  global_norm_kernel<<<Bn, 256, 0, stream>>>(wx, postsq, out);
}